// _LightGCNModel_31628139168294
// MI455X (gfx1250) — compile-verified
//
#include <hip/hip_runtime.h>

#define N_USERS 100000
#define N_ITEMS 50000
#define N_NODES 150000
#define N_EDGES 4000000
#define EMB 64
#define BATCH 16384

typedef __attribute__((ext_vector_type(2))) float v2f;
typedef __attribute__((ext_vector_type(8))) float v8f;

// ---------------------------------------------------------------------------
// Elementwise helpers (float4-wide over N_NODES*EMB = 9.6M floats)
// ---------------------------------------------------------------------------
__global__ __launch_bounds__(256) void zero_f4(float4* __restrict__ p, int n4) {
    int i = blockIdx.x * 256 + threadIdx.x;
    if (i < n4) p[i] = make_float4(0.f, 0.f, 0.f, 0.f);
}

__global__ __launch_bounds__(256) void add_f4(const float4* __restrict__ a,
                                              const float4* __restrict__ b,
                                              float4* __restrict__ o, int n4) {
    int i = blockIdx.x * 256 + threadIdx.x;
    if (i < n4) {
        float4 x = a[i], y = b[i];
        o[i] = make_float4(x.x + y.x, x.y + y.y, x.z + y.z, x.w + y.w);
    }
}

// ---------------------------------------------------------------------------
// Scatter-atomic SpMM: E_out[row] += val * E_in[col].
// 16 threads per edge, one float4 feature chunk each. E table (38.4MB) is
// L2-resident on MI455X (192MB L2), so gathers and f32 atomics hit L2.
// ---------------------------------------------------------------------------
__global__ __launch_bounds__(256) void spmm_scatter(
    const float* __restrict__ E_in, float* __restrict__ E_out,
    const float* __restrict__ edge_val, const int* __restrict__ edge_row,
    const int* __restrict__ edge_col) {
    unsigned tid = blockIdx.x * 256u + threadIdx.x;
    unsigned e = tid >> 4;
    if (e >= N_EDGES) return;
    unsigned sub = (tid & 15u) * 4u;
    float val = edge_val[e];
    int c = edge_col[e];
    int r = edge_row[e];
    const float4 x = *(const float4*)(E_in + (size_t)c * EMB + sub);
    float* dst = E_out + (size_t)r * EMB + sub;
    atomicAdd(dst + 0, val * x.x);
    atomicAdd(dst + 1, val * x.y);
    atomicAdd(dst + 2, val * x.z);
    atomicAdd(dst + 3, val * x.w);
}

// ---------------------------------------------------------------------------
// Batched dot via V_WMMA_F32_16X16X4_F32: one wave computes 16 pairs as
// diag( U(16x64) * I(16x64)^T ), accumulated over 16 chained k=4 WMMA steps.
// A-frag (16x4, MxK):  lane L<16 -> row L,  k = {0,1}; L>=16 -> row L-16, k={2,3}
// B-frag (4x16, KxN):  mirrored -> identical per-lane indexing with N for M.
// C/D (16x16): lanes 0-15: N=lane, M=vgpr; lanes 16-31: N=lane-16, M=vgpr+8.
// ---------------------------------------------------------------------------
__global__ __launch_bounds__(256) void predict_wmma(
    const float* __restrict__ E_sum, const int* __restrict__ user_batch,
    const int* __restrict__ item_batch, float* __restrict__ out) {
    const int lane = threadIdx.x & 31;
    const int wave = (blockIdx.x * 256 + threadIdx.x) >> 5;
    const int base = wave * 16;          // 16 (user,item) pairs per wave
    const int row = lane & 15;
    const int khalf = (lane >> 4) * 2;   // 0 for lanes 0-15, 2 for lanes 16-31

    const float* Up = E_sum + (size_t)user_batch[base + row] * EMB;
    const float* Ip = E_sum + ((size_t)item_batch[base + row] + N_USERS) * EMB;

    v8f acc = {};
#pragma unroll
    for (int kc = 0; kc < EMB; kc += 4) {
        v2f a, b;
        a.x = Up[kc + khalf];
        a.y = Up[kc + khalf + 1];
        b.x = Ip[kc + khalf];
        b.y = Ip[kc + khalf + 1];
        // 8 args: (neg_a, A, neg_b, B, c_mod, C, reuse_a, reuse_b)
        acc = __builtin_amdgcn_wmma_f32_16x16x4_f32(false, a, false, b,
                                                    (short)0, acc, false, false);
    }

    float tmp[8];
#pragma unroll
    for (int j = 0; j < 8; ++j) tmp[j] = acc[j];

    // Diagonal extraction; fold E_final = E_sum/4 into dot: x (1/16).
    int m = -1;
    if (lane < 8)        m = lane;        // D[m][m] in vgpr m,   lanes 0..7
    else if (lane >= 24) m = lane - 16;   // D[m][m] in vgpr m-8, lanes 24..31
    if (m >= 0) out[base + m] = tmp[m & 7] * 0.0625f;
}

// ---------------------------------------------------------------------------
extern "C" void kernel_launch(void* const* d_in, const int* in_sizes, int n_in,
                              void* d_out, int out_size, void* d_ws, size_t ws_size,
                              hipStream_t stream) {
    (void)in_sizes; (void)n_in; (void)out_size; (void)ws_size;

    const float* E0       = (const float*)d_in[0];
    const float* edge_val = (const float*)d_in[1];
    const int*   edge_row = (const int*)d_in[2];
    const int*   edge_col = (const int*)d_in[3];
    const int*   ub       = (const int*)d_in[4];
    const int*   ib       = (const int*)d_in[5];
    float*       out      = (float*)d_out;

    const size_t NELEM = (size_t)N_NODES * EMB;   // 9.6M floats = 38.4MB
    float* bufA = (float*)d_ws;
    float* bufB = bufA + NELEM;
    float* bufS = bufB + NELEM;                   // total ws: 115.2MB

    const int n4    = (int)(NELEM / 4);
    const int gridE = (n4 + 255) / 256;
    const int gridS = (int)(((size_t)N_EDGES * 16 + 255) / 256);

    // hop 1: bufA = A_hat * E0 ; bufS = E0 + bufA
    zero_f4<<<gridE, 256, 0, stream>>>((float4*)bufA, n4);
    spmm_scatter<<<gridS, 256, 0, stream>>>(E0, bufA, edge_val, edge_row, edge_col);
    add_f4<<<gridE, 256, 0, stream>>>((const float4*)E0, (const float4*)bufA,
                                      (float4*)bufS, n4);
    // hop 2: bufB = A_hat * bufA ; bufS += bufB
    zero_f4<<<gridE, 256, 0, stream>>>((float4*)bufB, n4);
    spmm_scatter<<<gridS, 256, 0, stream>>>(bufA, bufB, edge_val, edge_row, edge_col);
    add_f4<<<gridE, 256, 0, stream>>>((const float4*)bufS, (const float4*)bufB,
                                      (float4*)bufS, n4);
    // hop 3: bufA = A_hat * bufB ; bufS += bufA
    zero_f4<<<gridE, 256, 0, stream>>>((float4*)bufA, n4);
    spmm_scatter<<<gridS, 256, 0, stream>>>(bufB, bufA, edge_val, edge_row, edge_col);
    add_f4<<<gridE, 256, 0, stream>>>((const float4*)bufS, (const float4*)bufA,
                                      (float4*)bufS, n4);

    // final: diag-dot of gathered rows, scaled by 1/(K+1)^2 = 1/16
    predict_wmma<<<BATCH / 128, 256, 0, stream>>>(bufS, ub, ib, out);
}